// AttentionPoolHead_5463198400623
// MI455X (gfx1250) — compile-verified
//
#include <hip/hip_runtime.h>
#include <hip/hip_bf16.h>

// ---------------------------------------------------------------------------
// AttentionPoolHead, restructured:
//   tokens = concat(cls, storage, patches)            (B, L=4101, D=1024)
//   kv = LN(tokens);  q is a single broadcast row -> fold q into Wk:
//     scores[b,h,l] = rstd*(x.Up[h] - mean*sUp[h]) + ch[h]
//   softmax over l; fold LN into V side:
//     wsum[b,h,i] = g[i]*(sum_l a'[b,h,l]*x[b,l,i] - c2[b,h]) + beta[i]
//     ctx[b,j]    = wsum[b, j/64, :] . Wv[j,:] + bv[j]
//   pooled = ctx @ Wo^T + bo ; LN ; @ Wp^T + bp
// Both heavy passes use V_WMMA_F32_16X16X4_F32 (fp32 exact).
// ---------------------------------------------------------------------------

constexpr int kD = 1024;
constexpr int kB = 64;
constexpr int kS = 4;
constexpr int kN = 4096;
constexpr int kH = 16;
constexpr int kL = 1 + kS + kN;   // 4101
constexpr int kC = 14;
constexpr float kScale = 0.125f;  // 1/sqrt(64)
constexpr float kEps = 1e-5f;
constexpr int kTiles = (kL + 15) / 16;  // 257
constexpr int TSTRIDE = 1036;           // 16B-aligned rows, bank-conflict-free

typedef __attribute__((ext_vector_type(2))) float v2f;
typedef __attribute__((ext_vector_type(4))) float v4f;
typedef __attribute__((ext_vector_type(8))) float v8f;

__device__ __forceinline__ v8f wmma_f32_k4(v2f a, v2f b, v8f c) {
  // D(16x16,f32) = A(16x4,f32) x B(4x16,f32) + C
  return __builtin_amdgcn_wmma_f32_16x16x4_f32(false, a, false, b, (short)0, c,
                                               false, false);
}

__device__ __forceinline__ const float* token_row(const float* cls,
                                                  const float* storage,
                                                  const float* patches,
                                                  int b, int l) {
  if (l == 0) return cls + (size_t)b * kD;
  if (l <= kS) return storage + ((size_t)b * kS + (l - 1)) * kD;
  return patches + ((size_t)b * kN + (l - 1 - kS)) * kD;
}

// --------------------------- tiny prep kernels -----------------------------

// qp[j] = query . Wq[j,:] + bq[j]   (grid 1024 blocks x 128 thr)
__launch_bounds__(128)
__global__ void k_qp(const float* __restrict__ query, const float* __restrict__ Wq,
                     const float* __restrict__ bq, float* __restrict__ qp) {
  __shared__ float red[128];
  const int j = blockIdx.x, t = threadIdx.x;
  const float* Wrow = Wq + (size_t)j * kD;
  float acc = 0.f;
  for (int i = t; i < kD; i += 128) acc += query[i] * Wrow[i];
  red[t] = acc; __syncthreads();
  for (int off = 64; off > 0; off >>= 1) {
    if (t < off) red[t] += red[t + off];
    __syncthreads();
  }
  if (t == 0) qp[j] = red[0] + bq[j];
}

// W1[h,i] = scale * sum_{j'<64} qp[h*64+j'] * Wk[h*64+j', i];  Up = g .* W1
__launch_bounds__(256)
__global__ void k_uprep(const float* __restrict__ qp, const float* __restrict__ Wk,
                        const float* __restrict__ g, float* __restrict__ Up,
                        float* __restrict__ W1) {
  const int h = blockIdx.x >> 2;
  const int i = (blockIdx.x & 3) * 256 + threadIdx.x;
  float acc = 0.f;
#pragma unroll 4
  for (int jp = 0; jp < 64; ++jp)
    acc += qp[h * 64 + jp] * Wk[((size_t)(h * 64 + jp)) * kD + i];
  float w1 = kScale * acc;
  W1[h * kD + i] = w1;
  Up[h * kD + i] = g[i] * w1;
}

// sUp[h] = sum_i Up[h,i];  ch[h] = sum_i beta[i]*W1[h,i] + scale*(qp_h.bk_h)
__launch_bounds__(256)
__global__ void k_consts(const float* __restrict__ qp, const float* __restrict__ bk,
                         const float* __restrict__ Up, const float* __restrict__ W1,
                         const float* __restrict__ be, float* __restrict__ sUp,
                         float* __restrict__ ch) {
  __shared__ float red[256];
  const int h = blockIdx.x, t = threadIdx.x;
  float su = 0.f, cb = 0.f;
  for (int i = t; i < kD; i += 256) {
    su += Up[h * kD + i];
    cb += be[i] * W1[h * kD + i];
  }
  red[t] = su; __syncthreads();
  for (int off = 128; off > 0; off >>= 1) { if (t < off) red[t] += red[t + off]; __syncthreads(); }
  float SU = red[0]; __syncthreads();
  red[t] = cb; __syncthreads();
  for (int off = 128; off > 0; off >>= 1) { if (t < off) red[t] += red[t + off]; __syncthreads(); }
  float CB = red[0]; __syncthreads();
  red[t] = (t < 64) ? qp[h * 64 + t] * bk[h * 64 + t] : 0.f;
  __syncthreads();
  for (int off = 128; off > 0; off >>= 1) { if (t < off) red[t] += red[t + off]; __syncthreads(); }
  if (t == 0) { sUp[h] = SU; ch[h] = CB + kScale * red[0]; }
}

// ----------------------- pass A: LN stats + scores (WMMA) -------------------
// grid (257 tiles, 64 batch) x 128 threads (4 waves). One 16-token tile/block.
__launch_bounds__(128)
__global__ void k_passA(const float* __restrict__ cls, const float* __restrict__ storage,
                        const float* __restrict__ patches, const float* __restrict__ Up,
                        const float* __restrict__ sUp, const float* __restrict__ ch,
                        float* __restrict__ mean_out, float* __restrict__ rstd_out,
                        float* __restrict__ scores) {
  __shared__ float xs[16 * TSTRIDE];     // token tile, padded rows
  __shared__ float red[4 * 32 * 8];      // cross-wave accumulator reduce
  __shared__ float s_mean[16], s_rstd[16];

  const int b = blockIdx.y;
  const int l0 = blockIdx.x * 16;
  const int tid = threadIdx.x;
  const int lane = tid & 31;
  const int wave = tid >> 5;

  // ---- phase 1: coalesced b128 load into LDS + per-token sum/sumsq ----
  {
    const int r = tid >> 3;   // token row 0..15
    const int c = tid & 7;    // 8 loader threads per row
    int l = l0 + r;
    int lc = l < kL ? l : kL - 1;
    const float* src = token_row(cls, storage, patches, b, lc);
    float* dst = xs + r * TSTRIDE;
    float s1 = 0.f, s2 = 0.f;
#pragma unroll
    for (int i = 0; i < 32; ++i) {
      int idx4 = i * 8 + c;             // float4 index within the row
      v4f v = *(const v4f*)(src + idx4 * 4);
      *(v4f*)(dst + idx4 * 4) = v;
      s1 += v.x + v.y + v.z + v.w;
      s2 += v.x * v.x + v.y * v.y + v.z * v.z + v.w * v.w;
    }
    red[tid * 2] = s1;
    red[tid * 2 + 1] = s2;
  }
  __syncthreads();
  if (tid < 16) {
    float s1 = 0.f, s2 = 0.f;
    for (int c = 0; c < 8; ++c) { s1 += red[(tid * 8 + c) * 2]; s2 += red[(tid * 8 + c) * 2 + 1]; }
    float m = s1 * (1.0f / kD);
    float var = s2 * (1.0f / kD) - m * m;
    float r = rsqrtf(var + kEps);
    s_mean[tid] = m; s_rstd[tid] = r;
    int l = l0 + tid;
    if (l < kL) {
      mean_out[(size_t)b * kL + l] = m;
      rstd_out[(size_t)b * kL + l] = r;
    }
  }
  __syncthreads();

  // ---- phase 2: D[h, tok] += A(Up 16xK) x B(x^T Kx16), K split over waves --
  const int mn = lane & 15;   // A: head row / B: token column
  const int hi = lane >> 4;   // K sub-pair selector
  const float* Arow = Up + (size_t)mn * kD;
  const float* Brow = xs + mn * TSTRIDE;
  const int kbase = wave * 256;
  v8f acc = {};
#pragma unroll 4
  for (int kk = 0; kk < 64; ++kk) {
    const int k0 = kbase + kk * 4;
    v2f a = *(const v2f*)(Arow + k0 + 2 * hi);   // {K, K+1} per lane half
    v2f bb = *(const v2f*)(Brow + k0 + 2 * hi);  // ds_load_b64 from LDS
    acc = wmma_f32_k4(a, bb, acc);
  }

  // ---- phase 3: cross-wave reduce + LN-folded epilogue ----
#pragma unroll
  for (int v = 0; v < 8; ++v) red[wave * 256 + lane * 8 + v] = acc[v];
  __syncthreads();
  if (wave == 0) {
#pragma unroll
    for (int v = 0; v < 8; ++v) {
      float raw = red[lane * 8 + v] + red[256 + lane * 8 + v] +
                  red[512 + lane * 8 + v] + red[768 + lane * 8 + v];
      int h = v + 8 * hi;          // C/D layout: VGPR v -> M = v (+8 hi half)
      int n = lane & 15;           // token within tile
      int l = l0 + n;
      if (l < kL) {
        float sc = s_rstd[n] * (raw - s_mean[n] * sUp[h]) + ch[h];
        scores[((size_t)(b * kH + h)) * kL + l] = sc;
      }
    }
  }
}

// ----------------- softmax + fold rstd into attention weights --------------
// grid (B*H) x 256. In-place: scores -> a' = softmax*rstd; c2 = sum a'*mean.
__launch_bounds__(256)
__global__ void k_softmax(float* __restrict__ scores, const float* __restrict__ mean,
                          const float* __restrict__ rstd, float* __restrict__ c2) {
  __shared__ float red[256];
  const int bh = blockIdx.x;
  const int b = bh >> 4;
  const int t = threadIdx.x;
  float* row = scores + (size_t)bh * kL;
  const float* mrow = mean + (size_t)b * kL;
  const float* rrow = rstd + (size_t)b * kL;

  float mx = -3.0e38f;
  for (int l = t; l < kL; l += 256) mx = fmaxf(mx, row[l]);
  red[t] = mx; __syncthreads();
  for (int off = 128; off > 0; off >>= 1) { if (t < off) red[t] = fmaxf(red[t], red[t + off]); __syncthreads(); }
  mx = red[0]; __syncthreads();

  float sum = 0.f;
  for (int l = t; l < kL; l += 256) sum += __expf(row[l] - mx);
  red[t] = sum; __syncthreads();
  for (int off = 128; off > 0; off >>= 1) { if (t < off) red[t] += red[t + off]; __syncthreads(); }
  float inv = 1.0f / red[0]; __syncthreads();

  float cacc = 0.f;
  for (int l = t; l < kL; l += 256) {
    float w = __expf(row[l] - mx) * inv;
    float ap = w * rrow[l];
    row[l] = ap;
    cacc += ap * mrow[l];
  }
  red[t] = cacc; __syncthreads();
  for (int off = 128; off > 0; off >>= 1) { if (t < off) red[t] += red[t + off]; __syncthreads(); }
  if (t == 0) c2[bh] = red[0];
}

// --------------- pass B: wsum[b,h,i] = sum_l a'[b,h,l]*x[b,l,i] (WMMA) -----
// grid (64 dim-tiles, 64 batch) x 128 threads (4 waves interleave K tokens).
// Generic (guarded, concat-aware) steps only for s in {0,1,1025}; the main
// loop is pure-patches with hoisted addressing + global_prefetch.
__device__ __forceinline__ void passb_generic_step(
    int s, int hi, int mn, int b, int i0,
    const float* __restrict__ cls, const float* __restrict__ storage,
    const float* __restrict__ patches, const float* __restrict__ arow, v8f& acc) {
  const int la = s * 4 + 2 * hi;
  const int la0 = la < kL ? la : kL - 1;
  const int la1 = la + 1 < kL ? la + 1 : kL - 1;
  float z0 = (la < kL) ? 1.f : 0.f;
  float z1 = (la + 1 < kL) ? 1.f : 0.f;
  v2f a; a.x = arow[la0] * z0; a.y = arow[la1] * z1;   // zero tail tokens
  const float* r0 = token_row(cls, storage, patches, b, la0);
  const float* r1 = token_row(cls, storage, patches, b, la1);
  v2f bb; bb.x = r0[i0 + mn]; bb.y = r1[i0 + mn];
  acc = wmma_f32_k4(a, bb, acc);
}

__launch_bounds__(128)
__global__ void k_passB(const float* __restrict__ cls, const float* __restrict__ storage,
                        const float* __restrict__ patches, const float* __restrict__ ap,
                        const float* __restrict__ c2, const float* __restrict__ g,
                        const float* __restrict__ be, float* __restrict__ wsum) {
  __shared__ float red[4 * 32 * 8];
  const int b = blockIdx.y;
  const int i0 = blockIdx.x * 16;
  const int tid = threadIdx.x;
  const int lane = tid & 31;
  const int wave = tid >> 5;
  const int mn = lane & 15;   // A: head, B: dim column
  const int hi = lane >> 4;

  const float* arow = ap + ((size_t)(b * kH + mn)) * kL;
  v8f acc = {};

  int s = wave;
  if (s < 2) {                       // tokens 0..7: cls/storage boundary
    passb_generic_step(s, hi, mn, b, i0, cls, storage, patches, arow, acc);
    s += 4;
  }
  // main loop: s in [2,1024] -> tokens 8..4099, all inside patches, in-bounds
  const float* pbase = patches + (size_t)b * kN * kD + i0 + mn;  // l -> +(l-5)*kD
  for (; s < 1025; s += 4) {
    const int la = s * 4 + 2 * hi;
    const float* px = pbase + (size_t)(la - 5) * kD;
    v2f a = *(const v2f*)(arow + la);         // pair {la, la+1}; only 4B-aligned for odd mn
    v2f bb; bb.x = px[0]; bb.y = px[kD];
    __builtin_prefetch(px + 16 * kD, 0, 3);   // global_prefetch: next iter rows
    acc = wmma_f32_k4(a, bb, acc);
  }
  if (s == 1025) {                   // tokens 4100..4103: ragged tail
    passb_generic_step(s, hi, mn, b, i0, cls, storage, patches, arow, acc);
  }

#pragma unroll
  for (int v = 0; v < 8; ++v) red[wave * 256 + lane * 8 + v] = acc[v];
  __syncthreads();
  if (wave == 0) {
    const int i = i0 + mn;
    const float gi = g[i], bei = be[i];
#pragma unroll
    for (int v = 0; v < 8; ++v) {
      float raw = red[lane * 8 + v] + red[256 + lane * 8 + v] +
                  red[512 + lane * 8 + v] + red[768 + lane * 8 + v];
      int h = v + 8 * hi;
      wsum[((size_t)(b * kH + h)) * kD + i] = gi * (raw - c2[b * kH + h]) + bei;
    }
  }
}

// ------------- ctx[b,j] = wsum[b, j/64, :] . Wv[j,:] + bv[j] ---------------
__launch_bounds__(256)
__global__ void k_ctx(const float* __restrict__ wsum, const float* __restrict__ Wv,
                      const float* __restrict__ bv, float* __restrict__ ctx) {
  __shared__ float red[256];
  const int b = blockIdx.y, h = blockIdx.x, t = threadIdx.x;
  const int jj = t & 63, seg = t >> 6;
  const int j = h * 64 + jj;
  const float* wrow = wsum + ((size_t)(b * kH + h)) * kD + seg * 256;
  const float* Wrow = Wv + (size_t)j * kD + seg * 256;
  float acc = 0.f;
  for (int i = 0; i < 256; ++i) acc += wrow[i] * Wrow[i];
  red[t] = acc; __syncthreads();
  if (t < 64)
    ctx[(size_t)b * kD + j] = red[jj] + red[64 + jj] + red[128 + jj] + red[192 + jj] + bv[j];
}

// ------------- pooled[b,j] = ctx[b,:] . Wo[j,:] + bo[j] --------------------
__launch_bounds__(256)
__global__ void k_outproj(const float* __restrict__ ctx, const float* __restrict__ Wo,
                          const float* __restrict__ bo, float* __restrict__ pooled) {
  __shared__ float red[256];
  const int b = blockIdx.y, jt = blockIdx.x, t = threadIdx.x;
  const int jj = t & 63, seg = t >> 6;
  const int j = jt * 64 + jj;
  const float* crow = ctx + (size_t)b * kD + seg * 256;
  const float* Wrow = Wo + (size_t)j * kD + seg * 256;
  float acc = 0.f;
  for (int i = 0; i < 256; ++i) acc += crow[i] * Wrow[i];
  red[t] = acc; __syncthreads();
  if (t < 64)
    pooled[(size_t)b * kD + j] = red[jj] + red[64 + jj] + red[128 + jj] + red[192 + jj] + bo[j];
}

// ------------- final: LN(pooled) @ Wp^T + bp -> out (B, 14) ----------------
__launch_bounds__(256)
__global__ void k_final(const float* __restrict__ pooled, const float* __restrict__ g,
                        const float* __restrict__ be, const float* __restrict__ Wp,
                        const float* __restrict__ bp, float* __restrict__ out) {
  __shared__ float red[256];
  __shared__ float xn[kD];
  const int b = blockIdx.x, t = threadIdx.x;
  const float* row = pooled + (size_t)b * kD;
  float s1 = 0.f, s2 = 0.f;
  for (int i = t; i < kD; i += 256) { float v = row[i]; s1 += v; s2 += v * v; }
  red[t] = s1; __syncthreads();
  for (int off = 128; off > 0; off >>= 1) { if (t < off) red[t] += red[t + off]; __syncthreads(); }
  float m = red[0] * (1.0f / kD); __syncthreads();
  red[t] = s2; __syncthreads();
  for (int off = 128; off > 0; off >>= 1) { if (t < off) red[t] += red[t + off]; __syncthreads(); }
  float var = red[0] * (1.0f / kD) - m * m;
  float r = rsqrtf(var + kEps);
  __syncthreads();
  for (int i = t; i < kD; i += 256) xn[i] = (row[i] - m) * r * g[i] + be[i];
  __syncthreads();
  for (int c = 0; c < kC; ++c) {
    float acc = 0.f;
    for (int i = t; i < kD; i += 256) acc += xn[i] * Wp[(size_t)c * kD + i];
    red[t] = acc; __syncthreads();
    for (int off = 128; off > 0; off >>= 1) { if (t < off) red[t] += red[t + off]; __syncthreads(); }
    if (t == 0) out[b * kC + c] = red[0] + bp[c];
    __syncthreads();
  }
}

// ---------------------------------------------------------------------------
extern "C" void kernel_launch(void* const* d_in, const int* in_sizes, int n_in,
                              void* d_out, int out_size, void* d_ws, size_t ws_size,
                              hipStream_t stream) {
  (void)in_sizes; (void)n_in; (void)out_size; (void)ws_size;
  const float* cls     = (const float*)d_in[0];
  const float* storage = (const float*)d_in[1];
  const float* patches = (const float*)d_in[2];
  const float* query   = (const float*)d_in[3];
  const float* g_kv    = (const float*)d_in[4];
  const float* b_kv    = (const float*)d_in[5];
  const float* Wq      = (const float*)d_in[6];
  const float* Wk      = (const float*)d_in[7];
  const float* Wv      = (const float*)d_in[8];
  const float* bq      = (const float*)d_in[9];
  const float* bk      = (const float*)d_in[10];
  const float* bv      = (const float*)d_in[11];
  const float* Wo      = (const float*)d_in[12];
  const float* bo      = (const float*)d_in[13];
  const float* g_out   = (const float*)d_in[14];
  const float* b_out   = (const float*)d_in[15];
  const float* Wp      = (const float*)d_in[16];
  const float* bp      = (const float*)d_in[17];
  float* out = (float*)d_out;
  float* ws  = (float*)d_ws;

  // workspace layout (floats)
  size_t o = 0;
  float* qp     = ws + o; o += 1024;
  float* Up     = ws + o; o += (size_t)kH * kD;
  float* W1     = ws + o; o += (size_t)kH * kD;
  float* sUp    = ws + o; o += 16;
  float* ch     = ws + o; o += 16;
  float* meanb  = ws + o; o += (size_t)kB * kL;
  float* rstdb  = ws + o; o += (size_t)kB * kL;
  float* c2     = ws + o; o += (size_t)kB * kH;
  float* scores = ws + o; o += (size_t)kB * kH * kL;
  float* wsum   = ws + o; o += (size_t)kB * kH * kD;
  float* ctx    = ws + o; o += (size_t)kB * kD;
  float* pooled = ws + o; o += (size_t)kB * kD;

  k_qp<<<dim3(kD), dim3(128), 0, stream>>>(query, Wq, bq, qp);
  k_uprep<<<dim3(kH * 4), dim3(256), 0, stream>>>(qp, Wk, g_kv, Up, W1);
  k_consts<<<dim3(kH), dim3(256), 0, stream>>>(qp, bk, Up, W1, b_kv, sUp, ch);
  k_passA<<<dim3(kTiles, kB), dim3(128), 0, stream>>>(cls, storage, patches, Up, sUp, ch,
                                                      meanb, rstdb, scores);
  k_softmax<<<dim3(kB * kH), dim3(256), 0, stream>>>(scores, meanb, rstdb, c2);
  k_passB<<<dim3(kD / 16, kB), dim3(128), 0, stream>>>(cls, storage, patches, scores, c2,
                                                       g_kv, b_kv, wsum);
  k_ctx<<<dim3(kH, kB), dim3(256), 0, stream>>>(wsum, Wv, bv, ctx);
  k_outproj<<<dim3(kH, kB), dim3(256), 0, stream>>>(ctx, Wo, bo, pooled);
  k_final<<<dim3(kB), dim3(256), 0, stream>>>(pooled, g_out, b_out, Wp, bp, out);
}